// PointNet2PartSegmentationSSG_712964571705
// MI455X (gfx1250) — compile-verified
//
#include <hip/hip_runtime.h>
#include <hip/hip_bf16.h>
#include <stdint.h>

// ---------------------------------------------------------------------------
// PointNet++ part segmentation (SSG) forward for gfx1250.
// All conv_bn_relu GEMMs run on V_WMMA_F32_16X16X4_F32 (fp32, matches ref).
// WMMA register mapping (CDNA5 ISA 7.12.2, 32-bit operands, wave32):
//   lane = half*16 + i   (half = lane>>4, i = lane&15)
//   A (16x4, M x K):  a[v] = A[i][2*half + v]
//   B (4x16, K x N):  b[v] = B[2*half + v][i]
//   C/D (16x16):      c[r] = D[r + 8*half][i]
// ---------------------------------------------------------------------------

#define BN_RS 0.9999950000374996f   // 1/sqrt(1 + 1e-5)  (EPS_BN fused scale)

typedef float v2f __attribute__((ext_vector_type(2)));
typedef float v8f __attribute__((ext_vector_type(8)));

static __device__ __forceinline__ v8f wmma4(v2f a, v2f b, v8f c) {
  // (neg_a, A, neg_b, B, c_mod, C, reuse_a, reuse_b)
  return __builtin_amdgcn_wmma_f32_16x16x4_f32(false, a, false, b, (short)0, c,
                                               false, false);
}

static __device__ __forceinline__ void argmax_combine(float& v, int& idx,
                                                      float v2, int i2) {
  if (v2 > v || (v2 == v && i2 < idx)) { v = v2; idx = i2; }
}

// ---------------------------------------------------------------------------
// Farthest point sampling. One block per batch. Emits far-index BEFORE the
// distance update (matches jax.lax.scan in reference). First-max tie-break.
// xyz: (B,3,N) channel-major.  fidx: (B,npoint).  new_xyz: (B,3,npoint).
// ---------------------------------------------------------------------------
__global__ __launch_bounds__(256) void fps_kernel(
    const float* __restrict__ xyz, int* __restrict__ fidx,
    float* __restrict__ nxyz, int N, int npoint) {
  __shared__ float sx[2048], sy[2048], sz[2048], sdist[2048];
  __shared__ float swv[8];
  __shared__ int swi[8];
  __shared__ int s_far;
  const int b = blockIdx.x, t = threadIdx.x;
  const int w = t >> 5, lane = t & 31;
  const float* xb = xyz + (size_t)b * 3 * N;
  for (int j = t; j < N; j += 256) {
    sx[j] = xb[j]; sy[j] = xb[N + j]; sz[j] = xb[2 * N + j];
    sdist[j] = 1e10f;
  }
  if (t == 0) s_far = 0;
  __syncthreads();
  for (int it = 0; it < npoint; ++it) {
    const int far = s_far;
    const float cx = sx[far], cy = sy[far], cz = sz[far];
    float bv = -1.0f; int bi = 0;
    for (int j = t; j < N; j += 256) {
      float dx = sx[j] - cx, dy = sy[j] - cy, dz = sz[j] - cz;
      float d = dx * dx + dy * dy + dz * dz;
      float m = fminf(sdist[j], d);
      sdist[j] = m;
      if (m > bv) { bv = m; bi = j; }   // ascending j + strict > => earliest
    }
    for (int off = 16; off > 0; off >>= 1) {
      float v2 = __shfl_xor(bv, off, 32);
      int   i2 = __shfl_xor(bi, off, 32);
      argmax_combine(bv, bi, v2, i2);
    }
    if (lane == 0) { swv[w] = bv; swi[w] = bi; }
    __syncthreads();
    if (t == 0) {
      float v = swv[0]; int idx = swi[0];
      for (int q = 1; q < 8; ++q) argmax_combine(v, idx, swv[q], swi[q]);
      fidx[(size_t)b * npoint + it] = far;
      nxyz[(size_t)b * 3 * npoint + it] = cx;
      nxyz[(size_t)b * 3 * npoint + npoint + it] = cy;
      nxyz[(size_t)b * 3 * npoint + 2 * npoint + it] = cz;
      s_far = idx;
    }
    __syncthreads();
  }
}

// ---------------------------------------------------------------------------
// Ball query: first `ns` in-radius indices in ascending order, padded with the
// first hit; all N-1 if no hit (matches reference's sort/clip semantics).
// ---------------------------------------------------------------------------
__global__ __launch_bounds__(256) void ball_query_kernel(
    const float* __restrict__ xyz, const float* __restrict__ nxyz,
    int* __restrict__ gidx, int N, int S, int ns, float r2, int total) {
  int e = blockIdx.x * blockDim.x + threadIdx.x;
  if (e >= total) return;
  int b = e / S, n = e % S;
  const float* xb = xyz + (size_t)b * 3 * N;
  float cx = nxyz[(size_t)b * 3 * S + n];
  float cy = nxyz[(size_t)b * 3 * S + S + n];
  float cz = nxyz[(size_t)b * 3 * S + 2 * S + n];
  int* out = gidx + ((size_t)b * S + n) * ns;
  int cnt = 0, first = N - 1;
  for (int j = 0; j < N && cnt < ns; ++j) {
    float dx = xb[j] - cx, dy = xb[N + j] - cy, dz = xb[2 * N + j] - cz;
    if (dx * dx + dy * dy + dz * dz <= r2) {
      if (cnt == 0) first = j;
      out[cnt++] = j;
    }
  }
  for (int k = cnt; k < ns; ++k) out[k] = first;
}

// ---------------------------------------------------------------------------
// SA1 fused: group=32 samples, MLP 3->64->64->128 + max-pool, all in LDS.
// Block = 256 threads (8 waves); one block per (group n, batch b).
// ---------------------------------------------------------------------------
__global__ __launch_bounds__(256) void sa1_kernel(
    const float* __restrict__ x,    // (B,3,2048)
    const int* __restrict__ gidx,   // (B,512,32)
    const float* __restrict__ nx,   // (B,3,512)
    const float* __restrict__ W1, const float* __restrict__ B1,
    const float* __restrict__ T1, const float* __restrict__ G1,
    const float* __restrict__ W2, const float* __restrict__ B2,
    const float* __restrict__ T2, const float* __restrict__ G2,
    const float* __restrict__ W3, const float* __restrict__ B3,
    const float* __restrict__ T3, const float* __restrict__ G3,
    float* __restrict__ out)        // (B,128,512)
{
  __shared__ float sX0[4 * 32];
  __shared__ float sY1[64 * 32];
  __shared__ float sY2[64 * 32];
  const int n = blockIdx.x, b = blockIdx.y;
  const int t = threadIdx.x;
  const int w = t >> 5, lane = t & 31, half = lane >> 4, li = lane & 15;

  if (t < 128) {
    int c = t >> 5, s = t & 31;
    float v = 0.f;
    if (c < 3) {
      int j = gidx[((size_t)b * 512 + n) * 32 + s];
      v = x[(size_t)b * 3 * 2048 + (size_t)c * 2048 + j] -
          nx[(size_t)b * 3 * 512 + (size_t)c * 512 + n];
    }
    sX0[c * 32 + s] = v;
  }
  __syncthreads();

  if (w < 4) {  // layer1: 64x3 (K padded to 4)
    int m = w * 16;
    v8f a0 = {}; v8f a1 = {};
    int c0 = 2 * half;
    v2f af, b0, b1v;
    af[0] = (c0 < 3) ? W1[(m + li) * 3 + c0] : 0.f;
    af[1] = (c0 + 1 < 3) ? W1[(m + li) * 3 + c0 + 1] : 0.f;
    b0[0] = sX0[c0 * 32 + li];       b0[1] = sX0[(c0 + 1) * 32 + li];
    b1v[0] = sX0[c0 * 32 + 16 + li]; b1v[1] = sX0[(c0 + 1) * 32 + 16 + li];
    a0 = wmma4(af, b0, a0);
    a1 = wmma4(af, b1v, a1);
#pragma unroll
    for (int r = 0; r < 8; ++r) {
      int o = m + r + 8 * half;
      float sc = G1[o] * BN_RS, bi = B1[o], bt = T1[o];
      sY1[o * 32 + li]      = fmaxf((a0[r] + bi) * sc + bt, 0.f);
      sY1[o * 32 + 16 + li] = fmaxf((a1[r] + bi) * sc + bt, 0.f);
    }
  }
  __syncthreads();

  if (w < 4) {  // layer2: 64x64
    int m = w * 16;
    v8f a0 = {}; v8f a1 = {};
    for (int k = 0; k < 64; k += 4) {
      int c0 = k + 2 * half;
      v2f af, b0, b1v;
      af[0] = W2[(m + li) * 64 + c0];
      af[1] = W2[(m + li) * 64 + c0 + 1];
      b0[0] = sY1[c0 * 32 + li];       b0[1] = sY1[(c0 + 1) * 32 + li];
      b1v[0] = sY1[c0 * 32 + 16 + li]; b1v[1] = sY1[(c0 + 1) * 32 + 16 + li];
      a0 = wmma4(af, b0, a0);
      a1 = wmma4(af, b1v, a1);
    }
#pragma unroll
    for (int r = 0; r < 8; ++r) {
      int o = m + r + 8 * half;
      float sc = G2[o] * BN_RS, bi = B2[o], bt = T2[o];
      sY2[o * 32 + li]      = fmaxf((a0[r] + bi) * sc + bt, 0.f);
      sY2[o * 32 + 16 + li] = fmaxf((a1[r] + bi) * sc + bt, 0.f);
    }
  }
  __syncthreads();

  {  // layer3: 128x64 + max over 32 samples
    int m = w * 16;
    v8f a0 = {}; v8f a1 = {};
    for (int k = 0; k < 64; k += 4) {
      int c0 = k + 2 * half;
      v2f af, b0, b1v;
      af[0] = W3[(m + li) * 64 + c0];
      af[1] = W3[(m + li) * 64 + c0 + 1];
      b0[0] = sY2[c0 * 32 + li];       b0[1] = sY2[(c0 + 1) * 32 + li];
      b1v[0] = sY2[c0 * 32 + 16 + li]; b1v[1] = sY2[(c0 + 1) * 32 + 16 + li];
      a0 = wmma4(af, b0, a0);
      a1 = wmma4(af, b1v, a1);
    }
    float mx[8];
#pragma unroll
    for (int r = 0; r < 8; ++r) {
      int o = m + r + 8 * half;
      float sc = G3[o] * BN_RS, bi = B3[o], bt = T3[o];
      float y0 = fmaxf((a0[r] + bi) * sc + bt, 0.f);
      float y1 = fmaxf((a1[r] + bi) * sc + bt, 0.f);
      mx[r] = fmaxf(y0, y1);
    }
#pragma unroll
    for (int off = 1; off < 16; off <<= 1)
#pragma unroll
      for (int r = 0; r < 8; ++r)
        mx[r] = fmaxf(mx[r], __shfl_xor(mx[r], off, 32));
    if (li == 0) {
#pragma unroll
      for (int r = 0; r < 8; ++r) {
        int o = m + r + 8 * half;
        out[(size_t)b * 128 * 512 + (size_t)o * 512 + n] = mx[r];
      }
    }
  }
}

// ---------------------------------------------------------------------------
// SA2 fused: group=64 samples processed as 2 half-groups of 32 (LDS budget),
// MLP 131->128->128->256 + partial max-pool -> part(B,256,128,2).
// ---------------------------------------------------------------------------
__global__ __launch_bounds__(256) void sa2_kernel(
    const float* __restrict__ xyz,  // (B,3,512)  l1 xyz
    const float* __restrict__ pts,  // (B,128,512) l1 feats
    const int* __restrict__ gidx,   // (B,128,64)
    const float* __restrict__ nx,   // (B,3,128)
    const float* __restrict__ W1, const float* __restrict__ B1,
    const float* __restrict__ T1, const float* __restrict__ G1,
    const float* __restrict__ W2, const float* __restrict__ B2,
    const float* __restrict__ T2, const float* __restrict__ G2,
    const float* __restrict__ W3, const float* __restrict__ B3,
    const float* __restrict__ T3, const float* __restrict__ G3,
    float* __restrict__ part)       // (B,256,128,2)
{
  __shared__ float sX0[132 * 32];
  __shared__ float sY1[128 * 32];
  __shared__ float sY2[128 * 32];
  const int n = blockIdx.x >> 1, h = blockIdx.x & 1, b = blockIdx.y;
  const int t = threadIdx.x;
  const int w = t >> 5, lane = t & 31, half = lane >> 4, li = lane & 15;

  for (int e = t; e < 132 * 32; e += 256) {
    int c = e >> 5, s = e & 31;
    float v = 0.f;
    if (c < 131) {
      int j = gidx[((size_t)b * 128 + n) * 64 + h * 32 + s];
      if (c < 3)
        v = xyz[(size_t)b * 3 * 512 + (size_t)c * 512 + j] -
            nx[(size_t)b * 3 * 128 + (size_t)c * 128 + n];
      else
        v = pts[(size_t)b * 128 * 512 + (size_t)(c - 3) * 512 + j];
    }
    sX0[c * 32 + s] = v;
  }
  __syncthreads();

  {  // layer1: 128x131 (K padded to 132)
    int m = w * 16;
    v8f a0 = {}; v8f a1 = {};
    for (int k = 0; k < 132; k += 4) {
      int c0 = k + 2 * half;
      v2f af, b0, b1v;
      af[0] = (c0 < 131) ? W1[(size_t)(m + li) * 131 + c0] : 0.f;
      af[1] = (c0 + 1 < 131) ? W1[(size_t)(m + li) * 131 + c0 + 1] : 0.f;
      b0[0] = sX0[c0 * 32 + li];       b0[1] = sX0[(c0 + 1) * 32 + li];
      b1v[0] = sX0[c0 * 32 + 16 + li]; b1v[1] = sX0[(c0 + 1) * 32 + 16 + li];
      a0 = wmma4(af, b0, a0);
      a1 = wmma4(af, b1v, a1);
    }
#pragma unroll
    for (int r = 0; r < 8; ++r) {
      int o = m + r + 8 * half;
      float sc = G1[o] * BN_RS, bi = B1[o], bt = T1[o];
      sY1[o * 32 + li]      = fmaxf((a0[r] + bi) * sc + bt, 0.f);
      sY1[o * 32 + 16 + li] = fmaxf((a1[r] + bi) * sc + bt, 0.f);
    }
  }
  __syncthreads();

  {  // layer2: 128x128
    int m = w * 16;
    v8f a0 = {}; v8f a1 = {};
    for (int k = 0; k < 128; k += 4) {
      int c0 = k + 2 * half;
      v2f af, b0, b1v;
      af[0] = W2[(size_t)(m + li) * 128 + c0];
      af[1] = W2[(size_t)(m + li) * 128 + c0 + 1];
      b0[0] = sY1[c0 * 32 + li];       b0[1] = sY1[(c0 + 1) * 32 + li];
      b1v[0] = sY1[c0 * 32 + 16 + li]; b1v[1] = sY1[(c0 + 1) * 32 + 16 + li];
      a0 = wmma4(af, b0, a0);
      a1 = wmma4(af, b1v, a1);
    }
#pragma unroll
    for (int r = 0; r < 8; ++r) {
      int o = m + r + 8 * half;
      float sc = G2[o] * BN_RS, bi = B2[o], bt = T2[o];
      sY2[o * 32 + li]      = fmaxf((a0[r] + bi) * sc + bt, 0.f);
      sY2[o * 32 + 16 + li] = fmaxf((a1[r] + bi) * sc + bt, 0.f);
    }
  }
  __syncthreads();

  for (int mt = 0; mt < 2; ++mt) {  // layer3: 256x128, 2 M-tiles per wave
    int m = (w * 2 + mt) * 16;
    v8f a0 = {}; v8f a1 = {};
    for (int k = 0; k < 128; k += 4) {
      int c0 = k + 2 * half;
      v2f af, b0, b1v;
      af[0] = W3[(size_t)(m + li) * 128 + c0];
      af[1] = W3[(size_t)(m + li) * 128 + c0 + 1];
      b0[0] = sY2[c0 * 32 + li];       b0[1] = sY2[(c0 + 1) * 32 + li];
      b1v[0] = sY2[c0 * 32 + 16 + li]; b1v[1] = sY2[(c0 + 1) * 32 + 16 + li];
      a0 = wmma4(af, b0, a0);
      a1 = wmma4(af, b1v, a1);
    }
    float mx[8];
#pragma unroll
    for (int r = 0; r < 8; ++r) {
      int o = m + r + 8 * half;
      float sc = G3[o] * BN_RS, bi = B3[o], bt = T3[o];
      float y0 = fmaxf((a0[r] + bi) * sc + bt, 0.f);
      float y1 = fmaxf((a1[r] + bi) * sc + bt, 0.f);
      mx[r] = fmaxf(y0, y1);
    }
#pragma unroll
    for (int off = 1; off < 16; off <<= 1)
#pragma unroll
      for (int r = 0; r < 8; ++r)
        mx[r] = fmaxf(mx[r], __shfl_xor(mx[r], off, 32));
    if (li == 0) {
#pragma unroll
      for (int r = 0; r < 8; ++r) {
        int o = m + r + 8 * half;
        part[(((size_t)b * 256 + o) * 128 + n) * 2 + h] = mx[r];
      }
    }
  }
}

__global__ __launch_bounds__(256) void max2_kernel(
    const float* __restrict__ part, float* __restrict__ out, int total) {
  int e = blockIdx.x * blockDim.x + threadIdx.x;
  if (e >= total) return;
  out[e] = fmaxf(part[2 * (size_t)e], part[2 * (size_t)e + 1]);
}

// ---------------------------------------------------------------------------
// Generic WMMA GEMM + BN + ReLU:  Y(B,Cout,P) = act(W(Cout,Cin) @ X(B,CinPad,P))
// Block = 8 waves sharing one M-tile; W K-chunks staged through LDS (coalesced,
// 33-float padded stride -> conflict-free A-fragment reads). Wave w covers
// NT n-tiles starting at (blockIdx.y*8 + w)*16*NT; one A fragment feeds NT
// WMMAs. Requires P % (128*NT) == 0; CinPad % 4 == 0; pad rows of X zeroed.
// ---------------------------------------------------------------------------
template <int NT>
__global__ __launch_bounds__(256) void gemm_kernel(
    const float* __restrict__ W, const float* __restrict__ X,
    const float* __restrict__ bias, const float* __restrict__ gamma,
    const float* __restrict__ beta, float* __restrict__ Y,
    int Cout, int Cin, int CinPad, int P, int relu) {
  __shared__ float sW[16 * 33];   // 16 rows x 32-K chunk, stride 33
  const int b = blockIdx.z;
  const int t = threadIdx.x, w = t >> 5, lane = t & 31;
  const int half = lane >> 4, li = lane & 15;
  const int m = blockIdx.x * 16;
  const int n0 = (blockIdx.y * 8 + w) * 16 * NT;
  const float* Xb = X + (size_t)b * CinPad * P;
  v8f acc[NT];
#pragma unroll
  for (int nt = 0; nt < NT; ++nt) acc[nt] = {};

  for (int k0 = 0; k0 < CinPad; k0 += 32) {
    __syncthreads();   // previous chunk fully consumed
    {                   // cooperative, coalesced load of W[m:m+16][k0:k0+32]
      int e0 = t;       // 512 elements, 2 per thread
      int row = e0 >> 5, col = e0 & 31;
      int c = k0 + col;
      sW[row * 33 + col] =
          (m + row < Cout && c < Cin) ? W[(size_t)(m + row) * Cin + c] : 0.f;
      int e1 = t + 256;
      row = e1 >> 5; col = e1 & 31; c = k0 + col;
      sW[row * 33 + col] =
          (m + row < Cout && c < Cin) ? W[(size_t)(m + row) * Cin + c] : 0.f;
    }
    __syncthreads();
    const int kend = (CinPad - k0 < 32) ? (CinPad - k0) : 32;
    for (int kk = 0; kk < kend; kk += 4) {
      v2f af;
      af[0] = sW[li * 33 + kk + 2 * half];
      af[1] = sW[li * 33 + kk + 2 * half + 1];
      const int c0 = k0 + kk + 2 * half;
#pragma unroll
      for (int nt = 0; nt < NT; ++nt) {
        v2f bf;
        bf[0] = Xb[(size_t)c0 * P + n0 + nt * 16 + li];
        bf[1] = Xb[(size_t)(c0 + 1) * P + n0 + nt * 16 + li];
        acc[nt] = wmma4(af, bf, acc[nt]);
      }
    }
  }

  float* Yb = Y + (size_t)b * Cout * P;
#pragma unroll
  for (int r = 0; r < 8; ++r) {
    int o = m + r + 8 * half;
    if (o < Cout) {
      float bi = bias ? bias[o] : 0.f;
      float sc = gamma ? (gamma[o] * BN_RS) : 1.f;
      float bt = (gamma && beta) ? beta[o] : 0.f;
#pragma unroll
      for (int nt = 0; nt < NT; ++nt) {
        float v = acc[nt][r] + bi;
        if (gamma) v = v * sc + bt;
        if (relu) v = fmaxf(v, 0.f);
        Yb[(size_t)o * P + n0 + nt * 16 + li] = v;
      }
    }
  }
}

// ---------------------------------------------------------------------------
// Misc small kernels
// ---------------------------------------------------------------------------
__global__ __launch_bounds__(256) void maxpool_kernel(
    const float* __restrict__ X, float* __restrict__ Y, int P, int total) {
  int e = blockIdx.x * blockDim.x + threadIdx.x;
  if (e >= total) return;
  const float* xr = X + (size_t)e * P;
  float m = xr[0];
  for (int j = 1; j < P; ++j) m = fmaxf(m, xr[j]);
  Y[e] = m;
}

__global__ __launch_bounds__(256) void knn3_kernel(
    const float* __restrict__ xyz1, const float* __restrict__ xyz2,
    int* __restrict__ idx, float* __restrict__ wgt, int P1, int S, int total) {
  int e = blockIdx.x * blockDim.x + threadIdx.x;
  if (e >= total) return;
  int b = e / P1, n = e % P1;
  float cx = xyz1[(size_t)b * 3 * P1 + n];
  float cy = xyz1[(size_t)b * 3 * P1 + P1 + n];
  float cz = xyz1[(size_t)b * 3 * P1 + 2 * P1 + n];
  const float* x2 = xyz2 + (size_t)b * 3 * S;
  float d0 = 3.4e38f, d1 = 3.4e38f, d2 = 3.4e38f;
  int i0 = 0, i1 = 0, i2 = 0;
  for (int j = 0; j < S; ++j) {
    float dx = x2[j] - cx, dy = x2[S + j] - cy, dz = x2[2 * S + j] - cz;
    float d = dx * dx + dy * dy + dz * dz;
    if (d < d0)      { d2 = d1; i2 = i1; d1 = d0; i1 = i0; d0 = d; i0 = j; }
    else if (d < d1) { d2 = d1; i2 = i1; d1 = d; i1 = j; }
    else if (d < d2) { d2 = d; i2 = j; }
  }
  float w0 = 1.f / (d0 + 1e-8f), w1 = 1.f / (d1 + 1e-8f), w2 = 1.f / (d2 + 1e-8f);
  float ws = w0 + w1 + w2;
  size_t o = (size_t)e * 3;
  idx[o] = i0; idx[o + 1] = i1; idx[o + 2] = i2;
  wgt[o] = w0 / ws; wgt[o + 1] = w1 / ws; wgt[o + 2] = w2 / ws;
}

__global__ __launch_bounds__(256) void interp_kernel(
    float* __restrict__ dst, int dstB, const float* __restrict__ p2, int C,
    int S, const int* __restrict__ idx, const float* __restrict__ wgt, int P,
    int total) {
  int e = blockIdx.x * blockDim.x + threadIdx.x;
  if (e >= total) return;
  int b = e / (C * P), r = e % (C * P), c = r / P, n = r % P;
  const int* id = idx + ((size_t)b * P + n) * 3;
  const float* w = wgt + ((size_t)b * P + n) * 3;
  const float* p = p2 + ((size_t)b * C + c) * S;
  dst[(size_t)b * dstB + (size_t)c * P + n] =
      w[0] * p[id[0]] + w[1] * p[id[1]] + w[2] * p[id[2]];
}

__global__ __launch_bounds__(256) void copy_rows_kernel(
    float* __restrict__ dst, int dstB, const float* __restrict__ src, int srcB,
    int C, int P, int total) {
  int e = blockIdx.x * blockDim.x + threadIdx.x;
  if (e >= total) return;
  int b = e / (C * P), r = e % (C * P), c = r / P, n = r % P;
  dst[(size_t)b * dstB + (size_t)c * P + n] = src[(size_t)b * srcB + (size_t)c * P + n];
}

__global__ __launch_bounds__(256) void bcast_rows_kernel(
    float* __restrict__ dst, int dstB, const float* __restrict__ src, int srcB,
    int C, int P, int total) {
  int e = blockIdx.x * blockDim.x + threadIdx.x;
  if (e >= total) return;
  int b = e / (C * P), r = e % (C * P), c = r / P, n = r % P;
  dst[(size_t)b * dstB + (size_t)c * P + n] = src[(size_t)b * srcB + c];
}

// ---------------------------------------------------------------------------
// Host
// ---------------------------------------------------------------------------
static inline void gemm(const float* W, const float* X, const float* bias,
                        const float* gamma, const float* beta, float* Y,
                        int Cout, int Cin, int CinPad, int P, int relu, int B,
                        hipStream_t s) {
  if (P % 512 == 0) {
    dim3 grid((Cout + 15) / 16, P / 512, B);
    gemm_kernel<4><<<grid, 256, 0, s>>>(W, X, bias, gamma, beta, Y, Cout, Cin,
                                        CinPad, P, relu);
  } else {
    dim3 grid((Cout + 15) / 16, P / 128, B);
    gemm_kernel<1><<<grid, 256, 0, s>>>(W, X, bias, gamma, beta, Y, Cout, Cin,
                                        CinPad, P, relu);
  }
}

extern "C" void kernel_launch(void* const* d_in, const int* in_sizes, int n_in,
                              void* d_out, int out_size, void* d_ws,
                              size_t ws_size, hipStream_t stream) {
  (void)n_in; (void)out_size; (void)ws_size;
  const float* x = (const float*)d_in[0];    // (B,3,2048)
  const float* cls = (const float*)d_in[1];  // (B,16)
  const int B = in_sizes[0] / (3 * 2048);

  // params flattened via jax pytree order: dict keys sorted.
  // top: fp1,fp2,fp3,head1,head2,sa1,sa2,sa3; per block: W,b,beta,gamma.
#define IN(k) ((const float*)d_in[(k)])
  const float *fp1W0 = IN(2), *fp1b0 = IN(3), *fp1t0 = IN(4), *fp1g0 = IN(5);
  const float *fp1W1 = IN(6), *fp1b1 = IN(7), *fp1t1 = IN(8), *fp1g1 = IN(9);
  const float *fp1W2 = IN(10), *fp1b2 = IN(11), *fp1t2 = IN(12), *fp1g2 = IN(13);
  const float *fp2W0 = IN(14), *fp2b0 = IN(15), *fp2t0 = IN(16), *fp2g0 = IN(17);
  const float *fp2W1 = IN(18), *fp2b1 = IN(19), *fp2t1 = IN(20), *fp2g1 = IN(21);
  const float *fp3W0 = IN(22), *fp3b0 = IN(23), *fp3t0 = IN(24), *fp3g0 = IN(25);
  const float *fp3W1 = IN(26), *fp3b1 = IN(27), *fp3t1 = IN(28), *fp3g1 = IN(29);
  const float *h1W = IN(30), *h1b = IN(31), *h1t = IN(32), *h1g = IN(33);
  const float *h2W = IN(34), *h2b = IN(35);
  const float *sa1W0 = IN(36), *sa1b0 = IN(37), *sa1t0 = IN(38), *sa1g0 = IN(39);
  const float *sa1W1 = IN(40), *sa1b1 = IN(41), *sa1t1 = IN(42), *sa1g1 = IN(43);
  const float *sa1W2 = IN(44), *sa1b2 = IN(45), *sa1t2 = IN(46), *sa1g2 = IN(47);
  const float *sa2W0 = IN(48), *sa2b0 = IN(49), *sa2t0 = IN(50), *sa2g0 = IN(51);
  const float *sa2W1 = IN(52), *sa2b1 = IN(53), *sa2t1 = IN(54), *sa2g1 = IN(55);
  const float *sa2W2 = IN(56), *sa2b2 = IN(57), *sa2t2 = IN(58), *sa2g2 = IN(59);
  const float *sa3W0 = IN(60), *sa3b0 = IN(61), *sa3t0 = IN(62), *sa3g0 = IN(63);
  const float *sa3W1 = IN(64), *sa3b1 = IN(65), *sa3t1 = IN(66), *sa3g1 = IN(67);
  const float *sa3W2 = IN(68), *sa3b2 = IN(69), *sa3t2 = IN(70), *sa3g2 = IN(71);
#undef IN

  // --- workspace: persistent region + one stage-reused scratch blob ---
  char* base = (char*)d_ws;
  size_t off = 0;
  auto alloc = [&](size_t bytes) -> void* {
    void* p = base + off;
    off = (off + bytes + 255) & ~(size_t)255;
    return p;
  };
  float* nxyz1 = (float*)alloc((size_t)B * 3 * 512 * 4);
  float* l1pts = (float*)alloc((size_t)B * 128 * 512 * 4);
  float* nxyz2 = (float*)alloc((size_t)B * 3 * 128 * 4);
  float* l2pts = (float*)alloc((size_t)B * 256 * 128 * 4);
  float* l3pts = (float*)alloc((size_t)B * 1024 * 4);
  float* l2new = (float*)alloc((size_t)B * 256 * 128 * 4);
  float* l1new = (float*)alloc((size_t)B * 128 * 512 * 4);
  char* scratch = base + off;
  size_t so = 0;
  auto sal = [&](size_t bytes) -> void* {
    void* p = scratch + so;
    so = (so + bytes + 255) & ~(size_t)255;
    return p;
  };

  // ======================= SA1 (stage A scratch) ==========================
  so = 0;
  int* fidx1 = (int*)sal((size_t)B * 512 * 4);
  int* gidx1 = (int*)sal((size_t)B * 512 * 32 * 4);
  fps_kernel<<<B, 256, 0, stream>>>(x, fidx1, nxyz1, 2048, 512);
  {
    int total = B * 512;
    ball_query_kernel<<<(total + 255) / 256, 256, 0, stream>>>(
        x, nxyz1, gidx1, 2048, 512, 32, 0.04f, total);
  }
  sa1_kernel<<<dim3(512, B), 256, 0, stream>>>(
      x, gidx1, nxyz1, sa1W0, sa1b0, sa1t0, sa1g0, sa1W1, sa1b1, sa1t1, sa1g1,
      sa1W2, sa1b2, sa1t2, sa1g2, l1pts);

  // ======================= SA2 (stage B scratch) ==========================
  so = 0;
  int* fidx2 = (int*)sal((size_t)B * 128 * 4);
  int* gidx2 = (int*)sal((size_t)B * 128 * 64 * 4);
  float* part2 = (float*)sal((size_t)B * 256 * 128 * 2 * 4);
  fps_kernel<<<B, 256, 0, stream>>>(nxyz1, fidx2, nxyz2, 512, 128);
  {
    int total = B * 128;
    ball_query_kernel<<<(total + 255) / 256, 256, 0, stream>>>(
        nxyz1, nxyz2, gidx2, 512, 128, 64, 0.16f, total);
  }
  sa2_kernel<<<dim3(256, B), 256, 0, stream>>>(
      nxyz1, l1pts, gidx2, nxyz2, sa2W0, sa2b0, sa2t0, sa2g0, sa2W1, sa2b1,
      sa2t1, sa2g1, sa2W2, sa2b2, sa2t2, sa2g2, part2);
  {
    int total = B * 256 * 128;
    max2_kernel<<<(total + 255) / 256, 256, 0, stream>>>(part2, l2pts, total);
  }

  // ======================= SA3 (stage C scratch) ==========================
  so = 0;
  float* sa3in = (float*)sal((size_t)B * 260 * 128 * 4);
  float* h3a = (float*)sal((size_t)B * 256 * 128 * 4);
  float* h3b = (float*)sal((size_t)B * 512 * 128 * 4);
  float* h3c = (float*)sal((size_t)B * 1024 * 128 * 4);
  hipMemsetAsync(sa3in, 0, (size_t)B * 260 * 128 * 4, stream);
  {
    int total = B * 3 * 128;
    copy_rows_kernel<<<(total + 255) / 256, 256, 0, stream>>>(
        sa3in, 260 * 128, nxyz2, 3 * 128, 3, 128, total);
    total = B * 256 * 128;
    copy_rows_kernel<<<(total + 255) / 256, 256, 0, stream>>>(
        sa3in + 3 * 128, 260 * 128, l2pts, 256 * 128, 256, 128, total);
  }
  gemm(sa3W0, sa3in, sa3b0, sa3g0, sa3t0, h3a, 256, 259, 260, 128, 1, B, stream);
  gemm(sa3W1, h3a, sa3b1, sa3g1, sa3t1, h3b, 512, 256, 256, 128, 1, B, stream);
  gemm(sa3W2, h3b, sa3b2, sa3g2, sa3t2, h3c, 1024, 512, 512, 128, 1, B, stream);
  {
    int total = B * 1024;
    maxpool_kernel<<<(total + 255) / 256, 256, 0, stream>>>(h3c, l3pts, 128, total);
  }

  // ======================= FP3 (stage D scratch) ==========================
  so = 0;
  float* fp3in = (float*)sal((size_t)B * 1280 * 128 * 4);
  float* fp3h = (float*)sal((size_t)B * 256 * 128 * 4);
  {
    int total = B * 256 * 128;
    copy_rows_kernel<<<(total + 255) / 256, 256, 0, stream>>>(
        fp3in, 1280 * 128, l2pts, 256 * 128, 256, 128, total);
    total = B * 1024 * 128;
    bcast_rows_kernel<<<(total + 255) / 256, 256, 0, stream>>>(
        fp3in + 256 * 128, 1280 * 128, l3pts, 1024, 1024, 128, total);
  }
  gemm(fp3W0, fp3in, fp3b0, fp3g0, fp3t0, fp3h, 256, 1280, 1280, 128, 1, B, stream);
  gemm(fp3W1, fp3h, fp3b1, fp3g1, fp3t1, l2new, 256, 256, 256, 128, 1, B, stream);

  // ======================= FP2 (stage E scratch) ==========================
  so = 0;
  int* kidx2 = (int*)sal((size_t)B * 512 * 3 * 4);
  float* kw2 = (float*)sal((size_t)B * 512 * 3 * 4);
  float* fp2in = (float*)sal((size_t)B * 384 * 512 * 4);
  float* fp2h = (float*)sal((size_t)B * 256 * 512 * 4);
  {
    int total = B * 512;
    knn3_kernel<<<(total + 255) / 256, 256, 0, stream>>>(nxyz1, nxyz2, kidx2,
                                                         kw2, 512, 128, total);
    total = B * 128 * 512;
    copy_rows_kernel<<<(total + 255) / 256, 256, 0, stream>>>(
        fp2in, 384 * 512, l1pts, 128 * 512, 128, 512, total);
    total = B * 256 * 512;
    interp_kernel<<<(total + 255) / 256, 256, 0, stream>>>(
        fp2in + 128 * 512, 384 * 512, l2new, 256, 128, kidx2, kw2, 512, total);
  }
  gemm(fp2W0, fp2in, fp2b0, fp2g0, fp2t0, fp2h, 256, 384, 384, 512, 1, B, stream);
  gemm(fp2W1, fp2h, fp2b1, fp2g1, fp2t1, l1new, 128, 256, 256, 512, 1, B, stream);

  // ================== FP1 + heads (stage F scratch) =======================
  so = 0;
  int* kidx1 = (int*)sal((size_t)B * 2048 * 3 * 4);
  float* kw1 = (float*)sal((size_t)B * 2048 * 3 * 4);
  float* fp1in = (float*)sal((size_t)B * 148 * 2048 * 4);
  float* bufP = (float*)sal((size_t)B * 128 * 2048 * 4);
  float* bufQ = (float*)sal((size_t)B * 128 * 2048 * 4);
  hipMemsetAsync(fp1in, 0, (size_t)B * 148 * 2048 * 4, stream);
  {
    int total = B * 2048;
    knn3_kernel<<<(total + 255) / 256, 256, 0, stream>>>(x, nxyz1, kidx1, kw1,
                                                         2048, 512, total);
    total = B * 3 * 2048;
    copy_rows_kernel<<<(total + 255) / 256, 256, 0, stream>>>(
        fp1in, 148 * 2048, x, 3 * 2048, 3, 2048, total);
    total = B * 16 * 2048;
    bcast_rows_kernel<<<(total + 255) / 256, 256, 0, stream>>>(
        fp1in + 3 * 2048, 148 * 2048, cls, 16, 16, 2048, total);
    total = B * 128 * 2048;
    interp_kernel<<<(total + 255) / 256, 256, 0, stream>>>(
        fp1in + 19 * 2048, 148 * 2048, l1new, 128, 512, kidx1, kw1, 2048, total);
  }
  gemm(fp1W0, fp1in, fp1b0, fp1g0, fp1t0, bufP, 128, 147, 148, 2048, 1, B, stream);
  gemm(fp1W1, bufP, fp1b1, fp1g1, fp1t1, bufQ, 128, 128, 128, 2048, 1, B, stream);
  gemm(fp1W2, bufQ, fp1b2, fp1g2, fp1t2, bufP, 128, 128, 128, 2048, 1, B, stream);
  gemm(h1W, bufP, h1b, h1g, h1t, bufQ, 128, 128, 128, 2048, 1, B, stream);
  gemm(h2W, bufQ, h2b, nullptr, nullptr, (float*)d_out, 50, 128, 128, 2048, 0,
       B, stream);
}